// Pool2D_12094627905543
// MI455X (gfx1250) — compile-verified
//
#include <hip/hip_runtime.h>

// Max-pool 2x2 stride 2, NCHW fp32: (32,128,224,224) -> (32,128,112,112)
// Pure HBM-bandwidth-bound streaming kernel (~1.03 GB traffic, ~44us floor
// at 23.3 TB/s). Widest VMEM ops + non-temporal cache hints (gfx1250 TH=NT).

typedef float v4f __attribute__((ext_vector_type(4)));

#define DIM_N  32
#define DIM_C  128
#define DIM_H  224
#define DIM_W  224
#define DIM_OH 112
#define DIM_OW 112

// Each thread consumes 8 input columns of 2 input rows (4 x b128 loads, 64B)
// and produces 4 output pixels (1 x b128 store, 16B).
#define COLS_PER_THREAD 8
#define SEGS_PER_ROW    (DIM_W / COLS_PER_THREAD)          // 28
#define TOTAL_ROWS      (DIM_N * DIM_C * DIM_OH)            // 458752 (n,c,oh) rows
#define TOTAL_THREADS   ((unsigned)TOTAL_ROWS * SEGS_PER_ROW) // 12,845,056
#define BLOCK_SIZE      256
#define GRID_SIZE       (TOTAL_THREADS / BLOCK_SIZE)        // 50,176 exact

__global__ __launch_bounds__(BLOCK_SIZE) void
maxpool2x2_s2_kernel(const float* __restrict__ in, float* __restrict__ out) {
    const unsigned tid = blockIdx.x * BLOCK_SIZE + threadIdx.x;

    // Decompose: tid = row * SEGS_PER_ROW + seg ; row = nc * OH + oh
    const unsigned seg = tid % SEGS_PER_ROW;        // which 8-col input segment
    const unsigned row = tid / SEGS_PER_ROW;        // flattened (n*c, oh)
    const unsigned oh  = row % DIM_OH;
    const unsigned nc  = row / DIM_OH;

    // Input base: rows 2*oh and 2*oh+1, columns [8*seg, 8*seg+8)
    const size_t in_base = (size_t)nc * (DIM_H * DIM_W)
                         + (size_t)(2u * oh) * DIM_W
                         + (size_t)seg * COLS_PER_THREAD;
    const float* p0 = in + in_base;            // row 2*oh   (16B aligned)
    const float* p1 = p0 + DIM_W;              // row 2*oh+1 (16B aligned)

    // 4 x global_load_b128, non-temporal (stream: each byte read once)
    const v4f a0 = __builtin_nontemporal_load((const v4f*)(p0));
    const v4f a1 = __builtin_nontemporal_load((const v4f*)(p0 + 4));
    const v4f b0 = __builtin_nontemporal_load((const v4f*)(p1));
    const v4f b1 = __builtin_nontemporal_load((const v4f*)(p1 + 4));

    // 2x2 window max -> 4 outputs (v_max_num_f32)
    v4f o;
    o.x = fmaxf(fmaxf(a0.x, a0.y), fmaxf(b0.x, b0.y));
    o.y = fmaxf(fmaxf(a0.z, a0.w), fmaxf(b0.z, b0.w));
    o.z = fmaxf(fmaxf(a1.x, a1.y), fmaxf(b1.x, b1.y));
    o.w = fmaxf(fmaxf(a1.z, a1.w), fmaxf(b1.z, b1.w));

    // 1 x global_store_b128, non-temporal (16B aligned: seg*4 is mult of 4)
    const size_t out_base = (size_t)nc * (DIM_OH * DIM_OW)
                          + (size_t)oh * DIM_OW
                          + (size_t)seg * 4u;
    __builtin_nontemporal_store(o, (v4f*)(out + out_base));
}

extern "C" void kernel_launch(void* const* d_in, const int* in_sizes, int n_in,
                              void* d_out, int out_size, void* d_ws, size_t ws_size,
                              hipStream_t stream) {
    (void)in_sizes; (void)n_in; (void)d_ws; (void)ws_size; (void)out_size;
    const float* x = (const float*)d_in[0];
    float* y = (float*)d_out;
    maxpool2x2_s2_kernel<<<GRID_SIZE, BLOCK_SIZE, 0, stream>>>(x, y);
}